// ChebWavelet_63230508531930
// MI455X (gfx1250) — compile-verified
//
#include <hip/hip_runtime.h>

#define F_IN   64
#define F_OUT  128
#define KDIM   192   // K * F_IN = 3 * 64

typedef float v2f __attribute__((ext_vector_type(2)));
typedef float v8f __attribute__((ext_vector_type(8)));

// gfx1250 async global->LDS copy path (ASYNCcnt), with portable fallback.
#if defined(__has_builtin)
#  if __has_builtin(__builtin_amdgcn_global_load_async_to_lds_b128)
#    define HAS_ASYNC_LDS 1
#  endif
#endif
#ifndef HAS_ASYNC_LDS
#  define HAS_ASYNC_LDS 0
#endif

#if HAS_ASYNC_LDS
// Match the builtin's exact parameter type: gcc-vector int4 with HIP
// __device__ (addrspace 1) source and __shared__ (addrspace 3) destination.
typedef int async_b128_t __attribute__((vector_size(16)));
typedef __attribute__((address_space(1))) async_b128_t* gas_b128_p;
typedef __attribute__((address_space(3))) async_b128_t* las_b128_p;
#endif

__device__ __forceinline__ void async_wait_all() {
#if defined(__has_builtin)
#  if __has_builtin(__builtin_amdgcn_s_wait_asynccnt)
    __builtin_amdgcn_s_wait_asynccnt(0);
    return;
#  endif
#endif
    asm volatile("s_wait_asynccnt 0" ::: "memory");
}

// ---------------------------------------------------------------------------
// Zero-init workspace (T1 and S), float4 stores.
// ---------------------------------------------------------------------------
__global__ __launch_bounds__(256) void cheb_zero(float4* __restrict__ p, int n4) {
    int i = blockIdx.x * 256 + threadIdx.x;
    if (i < n4) p[i] = make_float4(0.f, 0.f, 0.f, 0.f);
}

// ---------------------------------------------------------------------------
// Fold the Chebyshev combine into the weights:
//   out = [X | T1 | 2S-X] @ W  ==  [X | T1 | S] @ W'   with
//   W'[0:64)   = W[0:64)   - W[128:192)
//   W'[64:128) = W[64:128)
//   W'[128:192)= 2 * W[128:192)
// ---------------------------------------------------------------------------
__global__ __launch_bounds__(256) void cheb_wprep(const float* __restrict__ W,
                                                  float* __restrict__ Wp) {
    int idx = blockIdx.x * 256 + threadIdx.x;
    if (idx >= KDIM * F_OUT) return;
    const int r = idx >> 7;               // row in [0,192)
    float v;
    if (r < F_IN)            v = W[idx] - W[idx + 2 * F_IN * F_OUT];
    else if (r < 2 * F_IN)   v = W[idx];
    else                     v = 2.f * W[idx];
    Wp[idx] = v;
}

// ---------------------------------------------------------------------------
// COO SpMM: out[row] += val * Y[col], one wave32 per edge.
// 32 lanes x float2 = one coalesced 256B gather per edge; scatter uses
// hardware FP32 global atomics (adds are L2-resident: Y is 16MB << 192MB L2).
// ---------------------------------------------------------------------------
__global__ __launch_bounds__(256) void cheb_spmm(
    const float* __restrict__ Y, const int* __restrict__ erow,
    const int* __restrict__ ecol, const float* __restrict__ eval,
    float* __restrict__ out, int nEdges) {
    const int eidx = blockIdx.x * 8 + (threadIdx.x >> 5);
    const int lane = threadIdx.x & 31;
    if (eidx >= nEdges) return;

    const int   r = erow[eidx];
    const int   c = ecol[eidx];
    const float v = eval[eidx];

    const float2 x = ((const float2*)(Y + (size_t)c * F_IN))[lane];
    float* dst = out + (size_t)r * F_IN + lane * 2;
    unsafeAtomicAdd(dst + 0, v * x.x);   // global_atomic_add_f32
    unsafeAtomicAdd(dst + 1, v * x.y);
}

// ---------------------------------------------------------------------------
// out[N,128] = [X | T1 | S] @ W'[192,128] + b   via V_WMMA_F32_16X16X4_F32.
// Block: 256 threads = 8 waves; block owns 32 rows x 128 cols. Each wave owns
// a 32x16 strip (two 16x16 accumulators sharing each B fragment -> half the
// W-load traffic per WMMA). H panel (32x192, pure copies now) is staged via
// async global->LDS b128 transfers; LDS row stride 196 floats keeps the
// A-fragment ds reads bank-conflict-free (banks 4*row + k, 64 distinct).
// ---------------------------------------------------------------------------
__global__ __launch_bounds__(256) void cheb_gemm_wmma(
    const float* __restrict__ X, const float* __restrict__ T1,
    const float* __restrict__ S, const float* __restrict__ Wp,
    const float* __restrict__ bias, float* __restrict__ out) {
    constexpr int LDH  = 196;             // padded LDS row stride (floats)
    constexpr int ROWS = 32;
    __shared__ __align__(16) float Hs[ROWS * LDH];

    const int m0  = blockIdx.x * ROWS;
    const int tid = threadIdx.x;

    // Stage H tile (32 x 192) into LDS: segment 0 = X, 1 = T1, 2 = S.
    for (int i = tid; i < ROWS * (KDIM / 4); i += 256) {
        const int r   = i / (KDIM / 4);   // row 0..31
        const int c4  = i % (KDIM / 4);   // float4 column 0..47
        const int seg = c4 >> 4;          // 64-feature segment
        const int f4  = c4 & 15;
        const float* src = (seg == 0) ? X : (seg == 1 ? T1 : S);
        const float4* g = (const float4*)src + (size_t)(m0 + r) * (F_IN / 4) + f4;
        float* l = &Hs[r * LDH + c4 * 4];
#if HAS_ASYNC_LDS
        __builtin_amdgcn_global_load_async_to_lds_b128(
            (gas_b128_p)g, (las_b128_p)l, /*offset=*/0, /*cpol=*/0);
#else
        *(float4*)l = *g;
#endif
    }
#if HAS_ASYNC_LDS
    async_wait_all();                     // s_wait_asynccnt 0
#endif
    __syncthreads();

    const int wave = tid >> 5;
    const int lane = tid & 31;
    const int n0   = wave * 16;
    const int lm   = lane & 15;           // A row / B,C column (ISA layout)
    const int kb   = (lane >> 4) * 2;     // K pair owned by this half-wave

    v8f acc0 = {}, acc1 = {};
    const float* wcol = Wp + n0 + lm;

    #pragma unroll 4
    for (int k0 = 0; k0 < KDIM; k0 += 4) {
        // B fragment: 4x16 f32, lane holds W'[k0+kb .. +1][n0+lm]
        v2f b;
        b.x = wcol[(size_t)(k0 + kb    ) * F_OUT];
        b.y = wcol[(size_t)(k0 + kb + 1) * F_OUT];
        // A fragments for the two stacked M-tiles
        v2f a0 = *(const v2f*)&Hs[ lm       * LDH + k0 + kb];
        v2f a1 = *(const v2f*)&Hs[(lm + 16) * LDH + k0 + kb];
        acc0 = __builtin_amdgcn_wmma_f32_16x16x4_f32(
            false, a0, false, b, (short)0, acc0, false, false);
        acc1 = __builtin_amdgcn_wmma_f32_16x16x4_f32(
            false, a1, false, b, (short)0, acc1, false, false);
    }

    // C/D 16x16 f32: VGPR v = row (v + 8*(lane>>4)), col lm.
    const int   rbase = (lane >> 4) * 8;
    const float bv    = bias[n0 + lm];
    float* op0 = out + (size_t)(m0 + rbase) * F_OUT + n0 + lm;
    float* op1 = op0 + (size_t)16 * F_OUT;
    #pragma unroll
    for (int v = 0; v < 8; ++v) op0[(size_t)v * F_OUT] = acc0[v] + bv;
    #pragma unroll
    for (int v = 0; v < 8; ++v) op1[(size_t)v * F_OUT] = acc1[v] + bv;
}

// ---------------------------------------------------------------------------
extern "C" void kernel_launch(void* const* d_in, const int* in_sizes, int n_in,
                              void* d_out, int out_size, void* d_ws, size_t ws_size,
                              hipStream_t stream) {
    const float* X    = (const float*)d_in[0];
    const int*   erow = (const int*)d_in[1];
    const int*   ecol = (const int*)d_in[2];
    const float* eval = (const float*)d_in[3];
    const float* W    = (const float*)d_in[4];
    const float* bias = (const float*)d_in[5];
    float*       out  = (float*)d_out;

    const int N = in_sizes[0] / F_IN;   // 65536 nodes
    const int E = in_sizes[1];          // 1048576 edges

    float* T1 = (float*)d_ws;                   // [N, 64]
    float* S  = T1 + (size_t)N * F_IN;          // [N, 64]
    float* Wp = S  + (size_t)N * F_IN;          // [192, 128] folded weights

    // Zero T1 and S (Wp is fully overwritten, no zeroing needed).
    {
        const int n4 = (N * F_IN * 2) / 4;
        cheb_zero<<<dim3((n4 + 255) / 256), dim3(256), 0, stream>>>((float4*)d_ws, n4);
    }
    // W' = [W0 - W2 ; W1 ; 2*W2]
    cheb_wprep<<<dim3((KDIM * F_OUT + 255) / 256), dim3(256), 0, stream>>>(W, Wp);
    // T1 = L @ X
    cheb_spmm<<<dim3((E + 7) / 8), dim3(256), 0, stream>>>(X, erow, ecol, eval, T1, E);
    // S  = L @ T1
    cheb_spmm<<<dim3((E + 7) / 8), dim3(256), 0, stream>>>(T1, erow, ecol, eval, S, E);
    // out = [X | T1 | S] @ W' + b
    cheb_gemm_wmma<<<dim3(N / 32), dim3(256), 0, stream>>>(X, T1, S, Wp, bias, out);
}